// Router_85718957294271
// MI455X (gfx1250) — compile-verified
//
#include <hip/hip_runtime.h>
#include <hip/hip_bf16.h>
#include <math.h>

// ---------------------------------------------------------------------------
// Router: B=524288 tokens, E=64 experts, D=32, H=4, HD=8, TOP_K=2.
// taskID has only 6 values -> everything up to clean_logits/noise_std is a
// function of t in [0,6). Precompute 6x64 tables (one wave, WMMA f32 16x16x4),
// then stream noise -> top2 -> gates with NT 128-bit memory ops (~270MB total,
// ~11.6us at 23.3 TB/s). load = column sums via deterministic two-level
// reduction (no float atomics).
// ---------------------------------------------------------------------------

#define RT_E 64
#define RT_D 32

typedef __attribute__((ext_vector_type(2))) float v2f;
typedef __attribute__((ext_vector_type(4))) float v4f;
typedef __attribute__((ext_vector_type(8))) float v8f;

// ============================ Kernel 1: precompute =========================
// One wave (32 lanes). Produces cleanTab[6][64], stdTab[6][64] in d_ws.
__global__ __launch_bounds__(32) void router_precompute(
    const float* __restrict__ embed_table,  // 6 x 32
    const float* __restrict__ expert_keys,  // 32 x 32
    const float* __restrict__ in_proj_w,    // 96 x 32 (Wq rows 0..31, Wk rows 32..63)
    const float* __restrict__ in_proj_b,    // 96
    const float* __restrict__ fc_gate_w,    // 64 x 32
    const float* __restrict__ fc_gate_b,    // 64
    const float* __restrict__ fc_noise_w,   // 64 x 32
    const float* __restrict__ fc_noise_b,   // 64
    float* __restrict__ cleanTab,           // 6 x 64
    float* __restrict__ stdTab)             // 6 x 64
{
    __shared__ float kbuf[32 * 32];   // K rows: [s][i]
    __shared__ float qbuf[6 * 32];    // Q rows: [t][i]
    __shared__ float sc[24 * 32];     // scores/probs: [(t*4+h)][s]
    __shared__ float ew[16 * 32];     // expert_weight padded to 16 rows (A operand)

    const int lane = threadIdx.x;     // 0..31 (wave32)

    // ---- K = expert_keys @ Wk^T + bk : lane s computes row s ----
    {
        const float* Wk = in_proj_w + RT_D * RT_D;
        const float* bk = in_proj_b + RT_D;
        const float* ek = expert_keys + lane * RT_D;
        for (int i = 0; i < RT_D; ++i) {
            float s = bk[i];
            const float* w = Wk + i * RT_D;
            for (int j = 0; j < RT_D; ++j) s += ek[j] * w[j];
            kbuf[lane * RT_D + i] = s;
        }
    }
    // ---- Q[t] = embed[t] @ Wq^T + bq : lanes 0..5 ----
    if (lane < 6) {
        const float* x = embed_table + lane * RT_D;
        for (int i = 0; i < RT_D; ++i) {
            float s = in_proj_b[i];
            const float* w = in_proj_w + i * RT_D;
            for (int j = 0; j < RT_D; ++j) s += x[j] * w[j];
            qbuf[lane * RT_D + i] = s;
        }
    }
    __syncthreads();

    // ---- scores[t][h][s] = <Q[t,h,:], K[s,h,:]> / sqrt(8) : lane = s ----
    const float inv_sqrt_hd = 0.35355339059327373f;
    for (int t = 0; t < 6; ++t)
        for (int h = 0; h < 4; ++h) {
            float s = 0.f;
            for (int d = 0; d < 8; ++d)
                s += qbuf[t * 32 + h * 8 + d] * kbuf[lane * 32 + h * 8 + d];
            sc[(t * 4 + h) * 32 + lane] = s * inv_sqrt_hd;
        }
    __syncthreads();

    // ---- softmax over s for each (t,h): one lane per pair (24 lanes) ----
    if (lane < 24) {
        float* row = &sc[lane * 32];
        float m = row[0];
        for (int s = 1; s < 32; ++s) m = fmaxf(m, row[s]);
        float sum = 0.f;
        for (int s = 0; s < 32; ++s) { float e = expf(row[s] - m); row[s] = e; sum += e; }
        float r = 1.f / sum;
        for (int s = 0; s < 32; ++s) row[s] *= r;
    }
    __syncthreads();

    // ---- zero-pad A operand rows 6..15 ----
    for (int i = lane; i < 16 * 32; i += 32) ew[i] = 0.f;
    __syncthreads();

    // ---- attn mean over h, then softmax over s -> expert_weight rows ----
    if (lane < 6) {
        float aw[32];
        for (int s = 0; s < 32; ++s)
            aw[s] = 0.25f * (sc[(lane * 4 + 0) * 32 + s] + sc[(lane * 4 + 1) * 32 + s] +
                             sc[(lane * 4 + 2) * 32 + s] + sc[(lane * 4 + 3) * 32 + s]);
        float m = aw[0];
        for (int s = 1; s < 32; ++s) m = fmaxf(m, aw[s]);
        float sum = 0.f;
        for (int s = 0; s < 32; ++s) { float e = expf(aw[s] - m); aw[s] = e; sum += e; }
        float r = 1.f / sum;
        for (int s = 0; s < 32; ++s) ew[lane * 32 + s] = aw[s] * r;
    }
    __syncthreads();

    // ---- [ew(16x32,padded)] x [G^T / N^T] via V_WMMA_F32_16X16X4_F32 ----
    // A 16x4 f32: lane m=lane&15 holds K = 2*(lane>>4) .. +1 in 2 VGPRs.
    // B 4x16 f32: lane n=lane&15 holds rows K = 2*(lane>>4) .. +1.
    // D 16x16 f32: VGPR v -> row v + 8*(lane>>4), col lane&15.
    const int m  = lane & 15;
    const int kk = (lane >> 4) * 2;
    const int rowBase = (lane >> 4) * 8;

    for (int nt = 0; nt < 4; ++nt) {
        v8f cg = {};
        v8f cn = {};
        const int n = nt * 16 + m;          // output column handled by this lane
#pragma unroll
        for (int kb = 0; kb < 8; ++kb) {    // K = 32 in steps of 4
            v2f a;
            a.x = ew[m * 32 + kb * 4 + kk];
            a.y = ew[m * 32 + kb * 4 + kk + 1];
            v2f bg;                          // B[k][n] = fc_gate_w[n][k]
            bg.x = fc_gate_w[n * 32 + kb * 4 + kk];
            bg.y = fc_gate_w[n * 32 + kb * 4 + kk + 1];
            v2f bn;
            bn.x = fc_noise_w[n * 32 + kb * 4 + kk];
            bn.y = fc_noise_w[n * 32 + kb * 4 + kk + 1];
            cg = __builtin_amdgcn_wmma_f32_16x16x4_f32(false, a, false, bg,
                                                       (short)0, cg, false, false);
            cn = __builtin_amdgcn_wmma_f32_16x16x4_f32(false, a, false, bn,
                                                       (short)0, cn, false, false);
        }
        // write back valid rows (t = 0..5)
#pragma unroll
        for (int v = 0; v < 8; ++v) {
            const int row = rowBase + v;
            if (row < 6) {
                cleanTab[row * 64 + n] = cg[v] + fc_gate_b[n];
                float z = cn[v] + fc_noise_b[n];
                // stable softplus: max(z,0) + log1p(exp(-|z|))
                float sp = fmaxf(z, 0.f) + log1pf(expf(-fabsf(z)));
                stdTab[row * 64 + n] = sp + 0.01f;   // NOISE_EPS
            }
        }
    }
}

// ============================ Kernel 2: streaming ==========================
// One token per thread. NT 128-bit loads of noise, NT 128-bit stores of gates.
// Deterministic per-block load partials (fixed-order LDS scan, no atomics).
__global__ __launch_bounds__(256) void router_main(
    const int*   __restrict__ taskID,
    const float* __restrict__ noise,     // B x 64
    const float* __restrict__ cleanTab,  // 6 x 64
    const float* __restrict__ stdTab,    // 6 x 64
    float* __restrict__ gates,           // B x 64
    float* __restrict__ partial,         // nblocks x 64
    int Btot)
{
    __shared__ float sClean[6 * 64];
    __shared__ float sStd[6 * 64];
    __shared__ int   sI0[256];
    __shared__ int   sI1[256];
    __shared__ float sG0[256];
    __shared__ float sG1[256];

    for (int i = threadIdx.x; i < 6 * 64; i += blockDim.x) {
        sClean[i] = cleanTab[i];
        sStd[i]   = stdTab[i];
    }
    __syncthreads();

    const int b = blockIdx.x * blockDim.x + threadIdx.x;
    int   i0 = -1, i1 = -1;
    float g0 = 0.f, g1 = 0.f;

    if (b < Btot) {
        const int t = taskID[b];
        const float* cl = &sClean[t * 64];
        const float* sd = &sStd[t * 64];
        const v4f* nz = (const v4f*)(noise + (size_t)b * 64);
        v4f* gz = (v4f*)(gates + (size_t)b * 64);

        float b0 = -3.4e38f, b1 = -3.4e38f;
        int j0 = 0, j1 = 0;
#pragma unroll
        for (int k = 0; k < 16; ++k) {
            v4f nv = __builtin_nontemporal_load(nz + k);   // streamed once
            float l;
            l = cl[4 * k + 0] + nv.x * sd[4 * k + 0];
            if (l > b0) { b1 = b0; j1 = j0; b0 = l; j0 = 4 * k + 0; } else if (l > b1) { b1 = l; j1 = 4 * k + 0; }
            l = cl[4 * k + 1] + nv.y * sd[4 * k + 1];
            if (l > b0) { b1 = b0; j1 = j0; b0 = l; j0 = 4 * k + 1; } else if (l > b1) { b1 = l; j1 = 4 * k + 1; }
            l = cl[4 * k + 2] + nv.z * sd[4 * k + 2];
            if (l > b0) { b1 = b0; j1 = j0; b0 = l; j0 = 4 * k + 2; } else if (l > b1) { b1 = l; j1 = 4 * k + 2; }
            l = cl[4 * k + 3] + nv.w * sd[4 * k + 3];
            if (l > b0) { b1 = b0; j1 = j0; b0 = l; j0 = 4 * k + 3; } else if (l > b1) { b1 = l; j1 = 4 * k + 3; }
        }
        // softmax over the (sorted) top-2
        g0 = 1.f / (1.f + __expf(b1 - b0));
        g1 = 1.f - g0;
        i0 = j0; i1 = j1;

        // write the gates row: two nonzeros, rest zero, one NT pass
#pragma unroll
        for (int k = 0; k < 16; ++k) {
            const int e = 4 * k;
            v4f v;
            v.x = (e + 0 == i0) ? g0 : ((e + 0 == i1) ? g1 : 0.f);
            v.y = (e + 1 == i0) ? g0 : ((e + 1 == i1) ? g1 : 0.f);
            v.z = (e + 2 == i0) ? g0 : ((e + 2 == i1) ? g1 : 0.f);
            v.w = (e + 3 == i0) ? g0 : ((e + 3 == i1) ? g1 : 0.f);
            __builtin_nontemporal_store(v, gz + k);
        }
    }

    // deterministic block-level load reduction (fixed scan order)
    sI0[threadIdx.x] = i0; sI1[threadIdx.x] = i1;
    sG0[threadIdx.x] = g0; sG1[threadIdx.x] = g1;
    __syncthreads();
    if (threadIdx.x < 64) {
        const int e = threadIdx.x;
        float acc = 0.f;
        for (int j = 0; j < 256; ++j) {
            acc += (sI0[j] == e) ? sG0[j] : 0.f;
            acc += (sI1[j] == e) ? sG1[j] : 0.f;
        }
        partial[(size_t)blockIdx.x * 64 + e] = acc;
    }
}

// ============================ Kernel 3: final reduce =======================
__global__ __launch_bounds__(64) void router_reduce(
    const float* __restrict__ partial, float* __restrict__ loadOut, int nblocks)
{
    const int e = threadIdx.x;   // 0..63
    float acc = 0.f;
    for (int blk = 0; blk < nblocks; ++blk)
        acc += partial[(size_t)blk * 64 + e];
    loadOut[e] = acc;
}

// ============================ Launch ======================================
extern "C" void kernel_launch(void* const* d_in, const int* in_sizes, int n_in,
                              void* d_out, int out_size, void* d_ws, size_t ws_size,
                              hipStream_t stream) {
    const int*   taskID      = (const int*)d_in[0];
    const float* embed_table = (const float*)d_in[1];
    const float* expert_keys = (const float*)d_in[2];
    const float* in_proj_w   = (const float*)d_in[3];
    const float* in_proj_b   = (const float*)d_in[4];
    const float* fc_gate_w   = (const float*)d_in[5];
    const float* fc_gate_b   = (const float*)d_in[6];
    const float* fc_noise_w  = (const float*)d_in[7];
    const float* fc_noise_b  = (const float*)d_in[8];
    const float* noise       = (const float*)d_in[9];

    const int Btot = in_sizes[0];                    // 524288
    float* gates   = (float*)d_out;                  // B x 64
    float* loadOut = gates + (size_t)Btot * 64;      // 64

    float* cleanTab = (float*)d_ws;                  // 6 x 64
    float* stdTab   = cleanTab + 6 * 64;             // 6 x 64
    float* partial  = stdTab + 6 * 64;               // nblocks x 64

    const int nblocks = (Btot + 255) / 256;          // 2048

    router_precompute<<<1, 32, 0, stream>>>(embed_table, expert_keys,
                                            in_proj_w, in_proj_b,
                                            fc_gate_w, fc_gate_b,
                                            fc_noise_w, fc_noise_b,
                                            cleanTab, stdTab);
    router_main<<<nblocks, 256, 0, stream>>>(taskID, noise, cleanTab, stdTab,
                                             gates, partial, Btot);
    router_reduce<<<1, 64, 0, stream>>>(partial, loadOut, nblocks);
}